// GCN_1752346657200
// MI455X (gfx1250) — compile-verified
//
#include <hip/hip_runtime.h>
#include <hip/hip_bf16.h>

typedef __attribute__((ext_vector_type(2))) float v2f;
typedef __attribute__((ext_vector_type(8))) float v8f;

#define N_NODES 50000
#define N_EDGES 600000
#define N_GRAPHS 64

// ---------------------------------------------------------------------------
// Degree accumulation: deg[col[e]] += 1  (self-loop +1 folded into rsqrt pass)
// ---------------------------------------------------------------------------
__global__ void gcn_degree(const long long* __restrict__ cols,
                           float* __restrict__ deg, int E) {
    int e = blockIdx.x * blockDim.x + threadIdx.x;
    if (e < E) unsafeAtomicAdd(&deg[cols[e]], 1.0f);
}

__global__ void gcn_rsqrt(float* __restrict__ dinv, int N) {
    int i = blockIdx.x * blockDim.x + threadIdx.x;
    if (i < N) dinv[i] = rsqrtf(dinv[i] + 1.0f);   // +1 self loop; always > 0
}

// ---------------------------------------------------------------------------
// Dense transform  Out[N x FOUT] = H[N x 128] @ W[128 x FOUT]
// One wave computes a 16 x (16*NT) strip: NT accumulators share one A
// fragment per K-step (4x A reuse).
// W is staged to LDS in K-pair interleaved layout:
//     sW[2*(p*FOUT + col) + r] = W[2p + r][col]   (r in {0,1})
// so each B fragment {W[k][col], W[k+1][col]} is ONE aligned ds_load_b64 at a
// compile-time immediate offset (no re-pairing movs in the WMMA loop).
// FOUT==64 path stages via CDNA5 async global->LDS (ASYNCcnt) then repacks;
// FOUT==128 repacks straight from global (raw+packed would exceed 64KB LDS).
// A 16x4 layout: lanes 0-15 hold K={0,1} in {v0,v1}; lanes 16-31 hold K={2,3}.
// C/D 16x16: VGPR v, lanes 0-15 -> M=v, lanes 16-31 -> M=v+8.
// ---------------------------------------------------------------------------
template <int FOUT, int NT>
__global__ void gcn_gemm_wmma(const float* __restrict__ H,
                              const float* __restrict__ W,
                              float* __restrict__ Out, int nStripes) {
    constexpr bool USE_ASYNC = (FOUT <= 64);
    __shared__ float sW[128 * FOUT];                    // interleaved pairs
    __shared__ float sRaw[USE_ASYNC ? 128 * FOUT : 4];  // async staging buffer

    const int tid = threadIdx.x;
    constexpr int CG = FOUT / 4;          // float4 groups per row
    constexpr int TASKS = 64 * CG;        // pair-rows x col-groups

    if constexpr (USE_ASYNC) {
        // CDNA5 async bulk copy: W -> sRaw (no VGPR round-trip)
        const float4* Wv  = (const float4*)W;
        float4*       sRv = (float4*)sRaw;
        for (int i = tid; i < (128 * FOUT) / 4; i += 256) {
            unsigned           lds_off = (unsigned)(uintptr_t)(sRv + i);
            unsigned long long gaddr   = (unsigned long long)(uintptr_t)(Wv + i);
            asm volatile("global_load_async_to_lds_b128 %0, %1, off"
                         :: "v"(lds_off), "v"(gaddr) : "memory");
        }
        asm volatile("s_wait_asynccnt 0" ::: "memory");
        __syncthreads();
        // repack sRaw (row-major) -> sW (K-pair interleaved)
        for (int i = tid; i < TASKS; i += 256) {
            int p = i / CG;
            int c = (i % CG) * 4;
            float4 r0 = *(const float4*)(sRaw + (size_t)(2 * p) * FOUT + c);
            float4 r1 = *(const float4*)(sRaw + (size_t)(2 * p + 1) * FOUT + c);
            float* dst = sW + 2 * ((size_t)p * FOUT + c);
            ((float4*)dst)[0] = make_float4(r0.x, r1.x, r0.y, r1.y);
            ((float4*)dst)[1] = make_float4(r0.z, r1.z, r0.w, r1.w);
        }
    } else {
        // direct repack: global W (L2-resident, 64KB) -> sW interleaved
        for (int i = tid; i < TASKS; i += 256) {
            int p = i / CG;
            int c = (i % CG) * 4;
            float4 r0 = *(const float4*)(W + (size_t)(2 * p) * FOUT + c);
            float4 r1 = *(const float4*)(W + (size_t)(2 * p + 1) * FOUT + c);
            float* dst = sW + 2 * ((size_t)p * FOUT + c);
            ((float4*)dst)[0] = make_float4(r0.x, r1.x, r0.y, r1.y);
            ((float4*)dst)[1] = make_float4(r0.z, r1.z, r0.w, r1.w);
        }
    }
    __syncthreads();

    const int wave = tid >> 5;
    const int lane = tid & 31;
    const int gw   = blockIdx.x * 8 + wave;

    constexpr int wavesPerStripe = FOUT / (16 * NT);   // 2 (FOUT=128), 1 (FOUT=64)
    const int stripe = gw / wavesPerStripe;            // 16-row stripe index
    const int nblk   = gw % wavesPerStripe;            // which 16*NT column chunk

    if (stripe < nStripes) {
        const int hi      = lane >> 4;                 // 0 or 1 (upper half-wave)
        const int mLoc    = lane & 15;
        const int colBase = nblk * (16 * NT) + mLoc;

        v8f acc[NT];
#pragma unroll
        for (int j = 0; j < NT; ++j) acc[j] = (v8f){};

        const float* hrow  = H + (size_t)(stripe * 16 + mLoc) * 128 + 2 * hi;
        const float* bbase = sW + 2 * ((size_t)hi * FOUT + colBase);

#pragma unroll
        for (int k0 = 0; k0 < 128; k0 += 4) {
            v2f a;
            a.x = hrow[k0];                            // H[row][k0 + 2*hi]
            a.y = hrow[k0 + 1];                        // H[row][k0 + 2*hi + 1]
#pragma unroll
            for (int j = 0; j < NT; ++j) {
                // {W[k0+2hi][col+16j], W[k0+2hi+1][col+16j]} : one ds_load_b64
                v2f b = *(const v2f*)(bbase + (size_t)k0 * FOUT + j * 32);
                acc[j] = __builtin_amdgcn_wmma_f32_16x16x4_f32(
                    false, a, false, b, (short)0, acc[j], false, false);
            }
        }

        float* obase = Out + ((size_t)stripe * 16 + 8 * hi) * FOUT + colBase;
#pragma unroll
        for (int j = 0; j < NT; ++j)
#pragma unroll
            for (int v = 0; v < 8; ++v)
                obase[(size_t)v * FOUT + j * 16] = acc[j][v];
    }
}

// ---------------------------------------------------------------------------
// Edge scatter: one wave per edge; lane owns F/32 features.
// out[d][f] += dinv[s]*dinv[d] * h[s][f]
// ---------------------------------------------------------------------------
template <int F>
__global__ void gcn_scatter(const long long* __restrict__ rows,
                            const long long* __restrict__ cols,
                            const float* __restrict__ dinv,
                            const float* __restrict__ h,
                            float* __restrict__ out, int E) {
    int wid  = (blockIdx.x * blockDim.x + threadIdx.x) >> 5;
    int lane = threadIdx.x & 31;
    if (wid >= E) return;
    long long s = rows[wid];
    long long d = cols[wid];
    float w = dinv[s] * dinv[d];
    constexpr int EL = F / 32;               // 4 (F=128) or 2 (F=64)
    const float* hp = h + (size_t)s * F + lane * EL;
    float*       op = out + (size_t)d * F + lane * EL;
#pragma unroll
    for (int j = 0; j < EL; ++j) unsafeAtomicAdd(op + j, w * hp[j]);
}

// ---------------------------------------------------------------------------
// Epilogue: agg = relu(agg + dinv[i]^2 * hT + bias)   (self loop + bias + relu)
// ---------------------------------------------------------------------------
template <int F>
__global__ void gcn_epilogue(float* __restrict__ agg,
                             const float* __restrict__ hT,
                             const float* __restrict__ dinv,
                             const float* __restrict__ bias, int N) {
    int idx = blockIdx.x * blockDim.x + threadIdx.x;   // one float4 per thread
    constexpr int C = F / 4;
    if (idx >= N * C) return;
    int i  = idx / C;
    int fc = (idx % C) * 4;
    float di = dinv[i];
    float w  = di * di;
    float4 a = *(const float4*)(agg + (size_t)i * F + fc);
    float4 t = *(const float4*)(hT + (size_t)i * F + fc);
    float4 b = *(const float4*)(bias + fc);
    float4 r;
    r.x = fmaxf(fmaf(w, t.x, a.x) + b.x, 0.0f);
    r.y = fmaxf(fmaf(w, t.y, a.y) + b.y, 0.0f);
    r.z = fmaxf(fmaf(w, t.z, a.z) + b.z, 0.0f);
    r.w = fmaxf(fmaf(w, t.w, a.w) + b.w, 0.0f);
    *(float4*)(agg + (size_t)i * F + fc) = r;
}

// ---------------------------------------------------------------------------
// Global mean pool (accumulate phase): sums[batch[i]][f] += h[i][f]
// ---------------------------------------------------------------------------
__global__ void gcn_pool(const float* __restrict__ h,
                         const long long* __restrict__ batch,
                         float* __restrict__ sums,
                         float* __restrict__ cnts, int N) {
    int idx = blockIdx.x * blockDim.x + threadIdx.x;
    if (idx >= N * 64) return;
    int i = idx >> 6;
    int f = idx & 63;
    long long g = batch[i];
    unsafeAtomicAdd(&sums[(size_t)g * 64 + f], h[(size_t)i * 64 + f]);
    if (f == 0) unsafeAtomicAdd(&cnts[g], 1.0f);
}

// ---------------------------------------------------------------------------
// Tiny MLP head: pooled[64x64] -> relu(@Wf1[64x32]+bf1) -> @Wf2[32x10]+bf2
// One thread per graph.
// ---------------------------------------------------------------------------
__global__ void gcn_mlp(const float* __restrict__ sums,
                        const float* __restrict__ cnts,
                        const float* __restrict__ Wf1,
                        const float* __restrict__ bf1,
                        const float* __restrict__ Wf2,
                        const float* __restrict__ bf2,
                        float* __restrict__ out) {
    int g = threadIdx.x;
    if (g >= N_GRAPHS) return;
    float inv = 1.0f / fmaxf(cnts[g], 1.0f);
    float p[64];
#pragma unroll
    for (int k = 0; k < 64; ++k) p[k] = sums[(size_t)g * 64 + k] * inv;
    float t1[32];
#pragma unroll 4
    for (int j = 0; j < 32; ++j) {
        float acc = bf1[j];
        for (int k = 0; k < 64; ++k) acc = fmaf(p[k], Wf1[k * 32 + j], acc);
        t1[j] = fmaxf(acc, 0.0f);
    }
#pragma unroll
    for (int o = 0; o < 10; ++o) {
        float acc = bf2[o];
        for (int j = 0; j < 32; ++j) acc = fmaf(t1[j], Wf2[j * 10 + o], acc);
        out[g * 10 + o] = acc;
    }
}

// ---------------------------------------------------------------------------
extern "C" void kernel_launch(void* const* d_in, const int* in_sizes, int n_in,
                              void* d_out, int out_size, void* d_ws, size_t ws_size,
                              hipStream_t stream) {
    const float*     x     = (const float*)d_in[0];
    const long long* ei    = (const long long*)d_in[1];   // [2, E] int64
    const long long* batch = (const long long*)d_in[2];
    const float* W1 = (const float*)d_in[3];  const float* b1 = (const float*)d_in[4];
    const float* W2 = (const float*)d_in[5];  const float* b2 = (const float*)d_in[6];
    const float* W3 = (const float*)d_in[7];  const float* b3 = (const float*)d_in[8];
    const float* Wf1 = (const float*)d_in[9];  const float* bf1 = (const float*)d_in[10];
    const float* Wf2 = (const float*)d_in[11]; const float* bf2 = (const float*)d_in[12];

    const int N = N_NODES, E = N_EDGES;
    const long long* erow = ei;        // edge_index[0] = source
    const long long* ecol = ei + E;    // edge_index[1] = target

    float* ws   = (float*)d_ws;
    float* dinv = ws;                                  // N
    float* bufA = ws + N;                              // N*128
    float* bufB = bufA + (size_t)N * 128;              // N*128
    float* psum = bufB + (size_t)N * 128;              // 64*64
    float* pcnt = psum + 64 * 64;                      // 64

    const int nStripes = N / 16;                       // 3125 (exact)
    // FOUT=128, NT=4: 2 waves/stripe -> 6250 waves -> 782 blocks of 8 waves
    const int gemmBlocks128 = (nStripes * 2 + 7) / 8;
    // FOUT=64,  NT=4: 1 wave/stripe  -> 3125 waves -> 391 blocks
    const int gemmBlocks64  = (nStripes + 7) / 8;

    // --- normalization coefficients -------------------------------------
    hipMemsetAsync(dinv, 0, (size_t)N * sizeof(float), stream);
    gcn_degree<<<(E + 255) / 256, 256, 0, stream>>>(ecol, dinv, E);
    gcn_rsqrt<<<(N + 255) / 256, 256, 0, stream>>>(dinv, N);

    // --- layer 1: x @ W1 -> aggregate -> relu(+b1) ----------------------
    gcn_gemm_wmma<128, 4><<<gemmBlocks128, 256, 0, stream>>>(x, W1, bufA, nStripes);
    hipMemsetAsync(bufB, 0, (size_t)N * 128 * sizeof(float), stream);
    gcn_scatter<128><<<(E + 7) / 8, 256, 0, stream>>>(erow, ecol, dinv, bufA, bufB, E);
    gcn_epilogue<128><<<(N * 32 + 255) / 256, 256, 0, stream>>>(bufB, bufA, dinv, b1, N);

    // --- layer 2 --------------------------------------------------------
    gcn_gemm_wmma<128, 4><<<gemmBlocks128, 256, 0, stream>>>(bufB, W2, bufA, nStripes);
    hipMemsetAsync(bufB, 0, (size_t)N * 128 * sizeof(float), stream);
    gcn_scatter<128><<<(E + 7) / 8, 256, 0, stream>>>(erow, ecol, dinv, bufA, bufB, E);
    gcn_epilogue<128><<<(N * 32 + 255) / 256, 256, 0, stream>>>(bufB, bufA, dinv, b2, N);

    // --- layer 3 (128 -> 64) --------------------------------------------
    gcn_gemm_wmma<64, 4><<<gemmBlocks64, 256, 0, stream>>>(bufB, W3, bufA, nStripes);
    hipMemsetAsync(bufB, 0, (size_t)N * 64 * sizeof(float), stream);
    gcn_scatter<64><<<(E + 7) / 8, 256, 0, stream>>>(erow, ecol, dinv, bufA, bufB, E);
    gcn_epilogue<64><<<(N * 16 + 255) / 256, 256, 0, stream>>>(bufB, bufA, dinv, b3, N);

    // --- mean pool + MLP head -------------------------------------------
    hipMemsetAsync(psum, 0, (64 * 64 + 64) * sizeof(float), stream);
    gcn_pool<<<(N * 64 + 255) / 256, 256, 0, stream>>>(bufB, batch, psum, pcnt, N);
    gcn_mlp<<<1, 64, 0, stream>>>(psum, pcnt, Wf1, bf1, Wf2, bf2, (float*)d_out);
}